// _RPN_FPN_19868518711726
// MI455X (gfx1250) — compile-verified
//
#include <hip/hip_runtime.h>
#include <math.h>
#include <stdint.h>

// ---------------------------------------------------------------------------
// RPN-FPN proposal network for MI455X (gfx1250).
// Conv 3x3 (256->512) as implicit GEMM on the fp32 WMMA pipe
// (V_WMMA_F32_16X16X4_F32; fp32 kept because the argmax-driven NMS is
// ordering-sensitive), with the B (weight) tile streamed into LDS by the
// Tensor Data Mover (tensor_load_to_lds + s_wait_tensorcnt) in a
// fragment-ready interleaved layout so every WMMA operand is one ds_load_b64.
// ---------------------------------------------------------------------------

typedef __attribute__((ext_vector_type(2))) float v2f;
typedef __attribute__((ext_vector_type(8))) float v8f;
typedef __attribute__((ext_vector_type(4))) unsigned int v4u;
typedef __attribute__((ext_vector_type(8))) int v8i;
typedef __attribute__((ext_vector_type(4))) int v4i;

// Problem constants
#define PPI   53294            // pixels per image (sum of level H*W)
#define NA    159882           // anchors per image (3 * PPI)
#define NB_   2                // batch
#define KTOT  2304             // 256 * 9 reduction length
#define COUT  512

// d_out layout (floats): rois | cls_prob | bbox_pred | anchors
#define OFF_CLS   8000
#define OFF_BBOX  327764
#define OFF_ANCH  1606820

// d_ws layout (float indices)
#define WT_F    0              // swizzled weights  [1152][1024] pair-interleaved
#define BOX_F   1179648        // clipped boxes     [2*NA][4]
#define SCR_F   2458704        // masked scores     [2*NA]
#define ISCR_F  2778468        // int region: hist[8192] cnt[2] cand[12288]
#define THR_F   2798950        // thresh[2] floats

#define CAND_CAP 6144
#define NEGV    -1e9f
#define SUPP    -1e10f

__device__ __forceinline__ void level_geom(int p, int& off, int& H, int& W,
                                           int& stride, float& scale) {
  if (p < 40000)      { off = 0;     H = 200; W = 200; stride = 4;  scale = 32.f;  }
  else if (p < 50000) { off = 40000; H = 100; W = 100; stride = 8;  scale = 64.f;  }
  else if (p < 52500) { off = 50000; H = 50;  W = 50;  stride = 16; scale = 128.f; }
  else if (p < 53125) { off = 52500; H = 25;  W = 25;  stride = 32; scale = 256.f; }
  else                { off = 53125; H = 13;  W = 13;  stride = 64; scale = 512.f; }
}

// --------------------------- weight pre-swizzle -----------------------------
// Fragment-ready, TDM-friendly layout:
//   Wt2[(k>>1)*1024 + co*2 + (k&1)] = conv_w[co*2304 + ci*9 + tap]
// where k = tap*256 + ci.  Each 16-k slab is a contiguous 32KB block.
__global__ void wtrans_kernel(const float* __restrict__ conv_w,
                              float* __restrict__ wt2) {
  int i = blockIdx.x * 256 + threadIdx.x;           // i = k*512 + co
  if (i >= KTOT * COUT) return;
  int k   = i >> 9;
  int co  = i & 511;
  int tap = k >> 8;
  int ci  = k & 255;
  wt2[((size_t)(k >> 1) << 10) + (co << 1) + (k & 1)] =
      conv_w[(size_t)co * KTOT + ci * 9 + tap];
}

// ------------------- fused conv3x3 + ReLU + heads + sigmoid -----------------
// One workgroup: 16 pixels x 512 channels.  8 waves, each 16x64 (4 WMMA tiles).
#define ASTR 18
#define BROW 1056              // 1024 dwords + TDM padding (8 dw per 256 dw)
#define XSTR 513               // epilogue x-staging stride (reuses Bs2)

__global__ __launch_bounds__(256)
void conv_head_kernel(const float* __restrict__ f0, const float* __restrict__ f1,
                      const float* __restrict__ f2, const float* __restrict__ f3,
                      const float* __restrict__ f4,
                      const float* __restrict__ wt2,
                      const float* __restrict__ conv_b,
                      const float* __restrict__ cls_w, const float* __restrict__ cls_b,
                      const float* __restrict__ bbox_w, const float* __restrict__ bbox_b,
                      float* __restrict__ out) {
  __shared__ float As[16 * ASTR];                 // [pixel][k] A tile
  __shared__ float Bs2[8 * BROW];                 // TDM-filled B tile; reused as x[pix][co]

  const int tid  = threadIdx.x;
  const int pix  = tid & 15;
  const int lane = tid & 31;
  const int wv   = tid >> 5;                      // wave id 0..7
  const int l    = lane & 15;
  const int h    = lane >> 4;

  // per-thread pixel geometry (pixel index is fixed = tid & 15)
  const int g   = blockIdx.x * 16 + pix;          // global pixel over B*PPI
  const bool gv = (g < NB_ * PPI);
  int bimg = 0, p = 0;
  if (gv) { bimg = g / PPI; p = g - bimg * PPI; }
  int off, H, W, stride; float scale;
  level_geom(p, off, H, W, stride, scale);
  const int y = (p - off) / W;
  const int x = (p - off) - y * W;
  const int HW = H * W;
  const float* fb;
  if (p < 40000)      fb = f0;
  else if (p < 50000) fb = f1;
  else if (p < 52500) fb = f2;
  else if (p < 53125) fb = f3;
  else                fb = f4;
  fb += (size_t)bimg * 256 * HW;

  const unsigned lds_b = (unsigned)(uintptr_t)(void*)&Bs2[0];  // LDS byte offset

  v8f acc[4];
  #pragma unroll
  for (int nb = 0; nb < 4; ++nb) acc[nb] = (v8f){0.f,0.f,0.f,0.f,0.f,0.f,0.f,0.f};

  for (int k0 = 0; k0 < KTOT; k0 += 16) {
    const int tap = k0 >> 8;
    const int ci0 = k0 & 255;
    const int dy  = tap / 3 - 1;
    const int dx  = tap % 3 - 1;
    __syncthreads();                              // previous compute done

    { // A tile: 16 pixels x 16 k, one element per thread
      const int kci = tid >> 4;
      const int yy = y + dy, xx = x + dx;
      float v = 0.f;
      if (gv && (unsigned)yy < (unsigned)H && (unsigned)xx < (unsigned)W)
        v = fb[(size_t)(ci0 + kci) * HW + yy * W + xx];
      As[pix * ASTR + kci] = v;
    }

    if (wv == 0) {
      // Tensor Data Mover: one 32KB contiguous slab -> LDS with padding
      // (8 dwords per 256) so h=0/h=1 fragment rows hit disjoint banks.
      const unsigned long long ga =
          (unsigned long long)(uintptr_t)(wt2 + ((size_t)(k0 >> 1) << 10));
      v4u g0;
      g0[0] = 1u;                                        // count=1, user mode
      g0[1] = lds_b;                                     // lds_addr (bytes)
      g0[2] = (unsigned)(ga & 0xffffffffu);              // global_addr lo
      g0[3] = (unsigned)((ga >> 32) & 0x01ffffffu) | (2u << 30);  // hi | type=2
      v8i g1;
      g1[0] = (int)((2u << 16)        // data_size = 4B
                  | (1u << 20)        // pad_enable
                  | (7u << 22)        // pad_interval = 256 dwords
                  | (7u << 25));      // pad_amount   = 8 dwords
      g1[1] = (int)0xFFFF0000u;       // tensor_dim0[15:0] in [31:16]
      g1[2] = (int)(0x7FFFu | (1u << 16));  // tensor_dim0 hi | tensor_dim1=1
      g1[3] = (int)(8192u << 16);     // tile_dim0 = 8192 elements
      g1[4] = 0;                      // tile_dim1 = tile_dim2 = 0 (unused)
      g1[5] = 0x40000000;             // tensor_dim0_stride (unused, 1 row)
      g1[6] = 0;
      g1[7] = 0;
      v4i gz4 = {0, 0, 0, 0};
      v8i gz8 = {0, 0, 0, 0, 0, 0, 0, 0};
      __builtin_amdgcn_tensor_load_to_lds(g0, g1, gz4, gz4, gz8, 0);
      __builtin_amdgcn_s_wait_tensorcnt(0);
    }
    __syncthreads();                              // tile visible to all waves

    // 4 WMMA K-steps of 4; every operand is a single aligned ds_load_b64
    #pragma unroll
    for (int kk = 0; kk < 4; ++kk) {
      const v2f a = *reinterpret_cast<const v2f*>(&As[l * ASTR + kk * 4 + 2 * h]);
      const int row = 2 * kk + h;
      #pragma unroll
      for (int nb = 0; nb < 4; ++nb) {
        const int col = (wv * 64 + nb * 16 + l) << 1;
        const v2f bf = *reinterpret_cast<const v2f*>(
            &Bs2[row * BROW + col + ((col >> 8) << 3)]);
        acc[nb] = __builtin_amdgcn_wmma_f32_16x16x4_f32(
            false, a, false, bf, (short)0, acc[nb], false, false);
      }
    }
  }

  // epilogue: x = relu(conv + bias) staged into the (now free) B buffer
  __syncthreads();
  #pragma unroll
  for (int nb = 0; nb < 4; ++nb) {
    const int ch = wv * 64 + nb * 16 + l;
    const float bias = conv_b[ch];
    #pragma unroll
    for (int r = 0; r < 8; ++r) {
      const int m = r + 8 * h;
      float xv = acc[nb][r] + bias;
      Bs2[m * XSTR + ch] = xv > 0.f ? xv : 0.f;
    }
  }
  __syncthreads();

  // heads: 16 pixels x 15 outputs (3 cls + 12 bbox), one dot(512) per thread
  if (tid < 240) {
    const int pp = tid / 15;
    const int o  = tid - pp * 15;
    const int gg = blockIdx.x * 16 + pp;
    if (gg < NB_ * PPI) {
      const float* wrow;
      float bias;
      if (o < 3) { wrow = cls_w  + o * 512;       bias = cls_b[o];      }
      else       { wrow = bbox_w + (o - 3) * 512; bias = bbox_b[o - 3]; }
      float s = bias;
      #pragma unroll 8
      for (int c = 0; c < 512; ++c) s += Bs2[pp * XSTR + c] * wrow[c];
      const int bb = gg / PPI;
      const int pl = gg - bb * PPI;
      if (o < 3) {
        const size_t row = (size_t)bb * NA + (size_t)pl * 3 + o;
        out[OFF_CLS + row] = 1.f / (1.f + expf(-s));
      } else {
        const int oo = o - 3;
        const size_t row = (size_t)bb * NA + (size_t)pl * 3 + (oo >> 2);
        out[OFF_BBOX + row * 4 + (oo & 3)] = s;
      }
    }
  }
}

// ------------------------------- anchors ------------------------------------
__global__ void anchors_kernel(float* __restrict__ out) {
  const int aid = blockIdx.x * 256 + threadIdx.x;
  if (aid >= NA) return;
  const int p = aid / 3;
  const int a = aid - 3 * p;
  int off, H, W, stride; float scale;
  level_geom(p, off, H, W, stride, scale);
  const int y = (p - off) / W;
  const int x = (p - off) - y * W;
  const float sr = (a == 0) ? 0.70710678118654752f
                 : (a == 1) ? 1.0f : 1.41421356237309505f;
  const float w  = scale * sr;
  const float hh = scale / sr;
  const float cx = (float)(x * stride);
  const float cy = (float)(y * stride);
  float* o = out + OFF_ANCH + (size_t)aid * 4;
  o[0] = cx - 0.5f * w;  o[1] = cy - 0.5f * hh;
  o[2] = cx + 0.5f * w;  o[3] = cy + 0.5f * hh;
}

// -------------------- decode + clip + filter + histogram --------------------
__global__ void decode_kernel(const float* __restrict__ im_info,
                              const float* __restrict__ out,
                              float* __restrict__ wsf,
                              int* __restrict__ hist) {
  const int i = blockIdx.x * 256 + threadIdx.x;
  if (i >= NB_ * NA) return;
  const int b   = i / NA;
  const int aid = i - b * NA;
  const float* d  = out + OFF_BBOX + (size_t)i * 4;
  const float* an = out + OFF_ANCH + (size_t)aid * 4;
  const float w  = an[2] - an[0] + 1.f;
  const float h  = an[3] - an[1] + 1.f;
  const float cx = an[0] + 0.5f * w;
  const float cy = an[1] + 0.5f * h;
  const float pcx = d[0] * w + cx;
  const float pcy = d[1] * h + cy;
  const float pw  = expf(d[2]) * w;
  const float ph  = expf(d[3]) * h;
  const float Hm1 = im_info[b * 3 + 0] - 1.f;
  const float Wm1 = im_info[b * 3 + 1] - 1.f;
  const float ms  = 16.f * im_info[b * 3 + 2];
  float x1 = fminf(fmaxf(pcx - 0.5f * pw, 0.f), Wm1);
  float y1 = fminf(fmaxf(pcy - 0.5f * ph, 0.f), Hm1);
  float x2 = fminf(fmaxf(pcx + 0.5f * pw, 0.f), Wm1);
  float y2 = fminf(fmaxf(pcy + 0.5f * ph, 0.f), Hm1);
  const bool valid = ((x2 - x1 + 1.f) >= ms) && ((y2 - y1 + 1.f) >= ms);
  const float prob = out[OFF_CLS + i];
  float* bx = wsf + BOX_F + (size_t)i * 4;
  bx[0] = x1; bx[1] = y1; bx[2] = x2; bx[3] = y2;
  wsf[SCR_F + i] = valid ? prob : NEGV;
  if (valid) {
    int bin = (int)(prob * 4096.f);
    bin = bin < 0 ? 0 : (bin > 4095 ? 4095 : bin);
    atomicAdd(&hist[b * 4096 + bin], 1);
  }
}

// ------------------------------ small kernels -------------------------------
__global__ void init_kernel(int* __restrict__ wsi) {
  const int i = blockIdx.x * 256 + threadIdx.x;
  if (i < 8192 + 2) wsi[i] = 0;                   // hist + cnt
}

__global__ void thresh_kernel(const int* __restrict__ hist,
                              float* __restrict__ thresh) {
  const int b = threadIdx.x;
  if (b < NB_) {
    int acc = 0, t = 0;
    for (int bin = 4095; bin >= 0; --bin) {
      acc += hist[b * 4096 + bin];
      if (acc >= 6000) { t = bin; break; }
    }
    thresh[b] = (float)t / 4096.f;
  }
}

__global__ void compact_kernel(const float* __restrict__ wsf,
                               const float* __restrict__ thresh,
                               int* __restrict__ cnt,
                               int* __restrict__ cand) {
  const int i = blockIdx.x * 256 + threadIdx.x;
  if (i >= NB_ * NA) return;
  const int b = i / NA;
  const float s = wsf[SCR_F + i];
  if (s >= thresh[b] && s > -1e8f) {
    const int pos = atomicAdd(&cnt[b], 1);
    if (pos < CAND_CAP) cand[b * CAND_CAP + pos] = i - b * NA;
  }
}

// ------------------------------ iterative NMS -------------------------------
__global__ __launch_bounds__(256)
void nms_kernel(const float* __restrict__ wsf,
                const int* __restrict__ cnt,
                const int* __restrict__ cand,
                float* __restrict__ out) {
  __shared__ float sc_s[CAND_CAP];
  __shared__ float red_v[256];
  __shared__ int   red_i[256];
  __shared__ float box_sh[4];
  const int b   = blockIdx.x;
  const int tid = threadIdx.x;
  const int n   = min(cnt[b], CAND_CAP);

  float4 bx[24];                                  // candidate boxes in VGPRs
  #pragma unroll
  for (int j = 0; j < 24; ++j) {
    const int slot = tid + j * 256;
    float s = -1e30f;
    float4 bb = make_float4(0.f, 0.f, 0.f, 0.f);
    if (slot < n) {
      const int aid = cand[b * CAND_CAP + slot];
      const size_t i = (size_t)b * NA + aid;
      const float* pb = wsf + BOX_F + i * 4;
      bb = make_float4(pb[0], pb[1], pb[2], pb[3]);
      s = wsf[SCR_F + i];
    }
    bx[j] = bb;
    sc_s[slot] = s;
  }
  __syncthreads();

  for (int it = 0; it < 1000; ++it) {
    float mv = -1e30f; int mi = tid;
    #pragma unroll
    for (int j = 0; j < 24; ++j) {
      const int slot = tid + j * 256;
      const float s = sc_s[slot];
      if (s > mv) { mv = s; mi = slot; }
    }
    red_v[tid] = mv; red_i[tid] = mi;
    __syncthreads();
    for (int st = 128; st > 0; st >>= 1) {
      if (tid < st) {
        const float ov = red_v[tid + st]; const int oi = red_i[tid + st];
        if (ov > red_v[tid] || (ov == red_v[tid] && oi < red_i[tid])) {
          red_v[tid] = ov; red_i[tid] = oi;
        }
      }
      __syncthreads();
    }
    const int wslot = red_i[0];
    if (tid == (wslot & 255)) {
      const int jw = wslot >> 8;
      float4 bb = bx[0];
      #pragma unroll
      for (int j = 1; j < 24; ++j) if (j == jw) bb = bx[j];
      box_sh[0] = bb.x; box_sh[1] = bb.y; box_sh[2] = bb.z; box_sh[3] = bb.w;
    }
    __syncthreads();
    const float X1 = box_sh[0], Y1 = box_sh[1], X2 = box_sh[2], Y2 = box_sh[3];
    if (tid == 0) {
      float* o = out + (size_t)b * 4000 + (size_t)it * 4;
      o[0] = X1; o[1] = Y1; o[2] = X2; o[3] = Y2;
    }
    const float a0 = (X2 - X1 + 1.f) * (Y2 - Y1 + 1.f);
    #pragma unroll
    for (int j = 0; j < 24; ++j) {
      const int slot = tid + j * 256;
      if (sc_s[slot] > -1e20f) {
        const float4 bb = bx[j];
        const float xx1 = fmaxf(X1, bb.x), yy1 = fmaxf(Y1, bb.y);
        const float xx2 = fminf(X2, bb.z), yy2 = fminf(Y2, bb.w);
        const float inter = fmaxf(xx2 - xx1 + 1.f, 0.f) * fmaxf(yy2 - yy1 + 1.f, 0.f);
        const float aa = (bb.z - bb.x + 1.f) * (bb.w - bb.y + 1.f);
        if (inter / (a0 + aa - inter) > 0.7f) sc_s[slot] = SUPP;
      }
    }
    __syncthreads();
  }
}

// --------------------------------- launch -----------------------------------
extern "C" void kernel_launch(void* const* d_in, const int* in_sizes, int n_in,
                              void* d_out, int out_size, void* d_ws, size_t ws_size,
                              hipStream_t stream) {
  (void)in_sizes; (void)n_in; (void)out_size; (void)ws_size;
  const float* f0      = (const float*)d_in[0];
  const float* f1      = (const float*)d_in[1];
  const float* f2      = (const float*)d_in[2];
  const float* f3      = (const float*)d_in[3];
  const float* f4      = (const float*)d_in[4];
  const float* im_info = (const float*)d_in[5];
  const float* conv_w  = (const float*)d_in[6];
  const float* conv_b  = (const float*)d_in[7];
  const float* cls_w   = (const float*)d_in[8];
  const float* cls_b   = (const float*)d_in[9];
  const float* bbox_w  = (const float*)d_in[10];
  const float* bbox_b  = (const float*)d_in[11];
  float* out = (float*)d_out;
  float* wsf = (float*)d_ws;
  int*   wsi = (int*)(wsf + ISCR_F);
  int*   hist   = wsi;
  int*   cnt    = wsi + 8192;
  int*   cand   = wsi + 8194;
  float* thresh = wsf + THR_F;

  init_kernel<<<(8194 + 255) / 256, 256, 0, stream>>>(wsi);
  wtrans_kernel<<<(KTOT * COUT) / 256, 256, 0, stream>>>(conv_w, wsf + WT_F);

  const int npix = NB_ * PPI;                     // 106588
  conv_head_kernel<<<(npix + 15) / 16, 256, 0, stream>>>(
      f0, f1, f2, f3, f4, wsf + WT_F, conv_b, cls_w, cls_b, bbox_w, bbox_b, out);

  anchors_kernel<<<(NA + 255) / 256, 256, 0, stream>>>(out);
  decode_kernel<<<(NB_ * NA + 255) / 256, 256, 0, stream>>>(im_info, out, wsf, hist);
  thresh_kernel<<<1, 64, 0, stream>>>(hist, thresh);
  compact_kernel<<<(NB_ * NA + 255) / 256, 256, 0, stream>>>(wsf, thresh, cnt, cand);
  nms_kernel<<<NB_, 256, 0, stream>>>(wsf, cnt, cand, out);
}